// get_model_3573412790741
// MI455X (gfx1250) — compile-verified
//
#include <hip/hip_runtime.h>
#include <limits.h>

typedef __attribute__((ext_vector_type(2))) float v2f;
typedef __attribute__((ext_vector_type(8))) float v8f;

#define NPTS   8192
#define NBATCH 64
#define S_DIM  384
#define GDIM   128
#define CHUNKS 4
#define PTS_PER_BLOCK (NPTS / CHUNKS)          // 2048
#define NTHREADS 256
#define NWAVES (NTHREADS / 32)                 // 8
#define GROUPS_PER_BLOCK (PTS_PER_BLOCK / 32)  // 64
#define GROUPS_PER_WAVE (GROUPS_PER_BLOCK / NWAVES) // 8

// Compute elevated = ELEVATE(3x3) * (pc*SCALE) for 32 points via two
// V_WMMA_F32_16X16X4_F32 ops. After merge, lane L owns point n0+L with
// e0,e1,e2 (elevated components).
__device__ __forceinline__ void elevate_group(const float* __restrict__ xb, int n0,
                                              int lane, float& e0, float& e1, float& e2)
{
    const float k = 0.4082482904638631f;  // 1/sqrt(6)
    const float SCALE = 190.0f;           // S//2 - 2
    const int m = lane & 15;
    const bool hi = lane >= 16;

    // A matrix (16x4 f32): lanes 0-15 hold (M=m, K=0/1) in v0/v1,
    // lanes 16-31 hold (M=m, K=2/3). Rows >=3 and K=3 are zero.
    float a0 = 0.f, a1 = 0.f;
    if (m < 3) {
        if (!hi) { a0 = (m == 0) ? 2.f * k : -k;  a1 = (m == 1) ? 2.f * k : -k; }
        else     { a0 = (m == 2) ? 2.f * k : -k;  a1 = 0.f; }
    }
    v2f A; A[0] = a0; A[1] = a1;

    // B matrix (4x16 f32): lanes 0-15 hold (K=0/1, N=m) in v0/v1,
    // lanes 16-31 hold (K=2/3, N=m). Branchless: pick the row-0/row-2 address
    // with a select, load row 1 unconditionally (always in-bounds) and mask it.
    const float* b0base = xb + (hi ? 2 * NPTS : 0);
    const int p = n0 + m;
    v2f B0, B1;
    {
        float r0a = b0base[p];
        float r1a = xb[NPTS + p];
        B0[0] = r0a * SCALE;
        B0[1] = hi ? 0.f : r1a * SCALE;
        float r0b = b0base[p + 16];
        float r1b = xb[NPTS + p + 16];
        B1[0] = r0b * SCALE;
        B1[1] = hi ? 0.f : r1b * SCALE;
    }
    v8f z = {0.f, 0.f, 0.f, 0.f, 0.f, 0.f, 0.f, 0.f};
    v8f dA = __builtin_amdgcn_wmma_f32_16x16x4_f32(false, A, false, B0, (short)0, z, false, false);
    v8f dB = __builtin_amdgcn_wmma_f32_16x16x4_f32(false, A, false, B1, (short)0, z, false, false);

    // D layout: VGPR r, lanes 0-15 = (M=r, N=lane). Rows 0..2 = elevated comps
    // for points n0..n0+15 (dA) and n0+16..n0+31 (dB). Move dB into upper lanes.
    int src = lane & 15;
    float u0 = __shfl(dB[0], src, 32);
    float u1 = __shfl(dB[1], src, 32);
    float u2 = __shfl(dB[2], src, 32);
    e0 = hi ? u0 : dA[0];
    e1 = hi ? u1 : dA[1];
    e2 = hi ? u2 : dA[2];
}

// Per-point lattice math: greedy round to multiple of 3 (correctly-rounded /3
// then round-half-even, matching jnp.round(elevated/3)), stable descending
// rank, remainder-sum correction. Returns adjusted greedy comps 0,1 and ranks 0,1.
__device__ __forceinline__ void lattice_point(float e0, float e1, float e2,
                                              int& gi0, int& gi1, int& ro0, int& ro1)
{
    float g0 = rintf(e0 / 3.0f) * 3.0f;
    float g1 = rintf(e1 / 3.0f) * 3.0f;
    float g2 = rintf(e2 / 3.0f) * 3.0f;
    float f0 = e0 - g0, f1 = e1 - g1, f2 = e2 - g2;
    // rank[i] = #{j: f[j] > f[i]} + #{j<i: f[j]==f[i]}  (stable descending argsort)
    int r0 = (int)(f1 > f0) + (int)(f2 > f0);
    int r1 = (int)(f0 >= f1) + (int)(f2 > f1);
    int rs = (int)rintf((g0 + g1 + g2) / 3.0f);  // exact integer, in {-1,0,1}
    if (rs > 0) {
        if (r0 >= 3 - rs) { g0 -= 3.f; r0 += rs - 3; } else r0 += rs;
        if (r1 >= 3 - rs) { g1 -= 3.f; r1 += rs - 3; } else r1 += rs;
    } else if (rs < 0) {
        if (r0 < -rs) { g0 += 3.f; r0 += rs + 3; } else r0 += rs;
        if (r1 < -rs) { g1 += 3.f; r1 += rs + 3; } else r1 += rs;
    }
    gi0 = (int)g0; gi1 = (int)g1; ro0 = r0; ro1 = r1;
}

// Kernel 0: zero the output, init per-batch offsets to INT_MAX.
__global__ __launch_bounds__(NTHREADS) void kinit(float4* __restrict__ out, int n4,
                                                  int* __restrict__ offs)
{
    int idx = blockIdx.x * blockDim.x + threadIdx.x;
    int stride = gridDim.x * blockDim.x;
    float4 z = {0.f, 0.f, 0.f, 0.f};
    for (int i = idx; i < n4; i += stride) out[i] = z;
    if (idx < 2 * NBATCH) offs[idx] = INT_MAX;
}

// Kernel 1: per-batch min over (g_c - rank_c), c = 0,1.
__global__ __launch_bounds__(NTHREADS) void kmin(const float* __restrict__ x,
                                                 int* __restrict__ offs)
{
    int b = blockIdx.x / CHUNKS, chunk = blockIdx.x % CHUNKS;
    const float* xb = x + (size_t)b * 6 * NPTS;
    int lane = threadIdx.x & 31, wave = threadIdx.x >> 5;
    int m0 = INT_MAX, m1 = INT_MAX;
    for (int t = 0; t < GROUPS_PER_WAVE; ++t) {
        int n0 = chunk * PTS_PER_BLOCK + (wave * GROUPS_PER_WAVE + t) * 32;
        float e0, e1, e2;
        elevate_group(xb, n0, lane, e0, e1, e2);
        int gi0, gi1, r0, r1;
        lattice_point(e0, e1, e2, gi0, gi1, r0, r1);
        m0 = min(m0, gi0 - r0);
        m1 = min(m1, gi1 - r1);
    }
    #pragma unroll
    for (int off = 16; off > 0; off >>= 1) {
        m0 = min(m0, __shfl_down(m0, off, 32));
        m1 = min(m1, __shfl_down(m1, off, 32));
    }
    __shared__ int s0[NWAVES], s1[NWAVES];
    if (lane == 0) { s0[wave] = m0; s1[wave] = m1; }
    __syncthreads();
    if (threadIdx.x == 0) {
        int a = s0[0], c = s1[0];
        for (int w = 1; w < NWAVES; ++w) { a = min(a, s0[w]); c = min(c, s1[w]); }
        atomicMin(&offs[2 * b], a);
        atomicMin(&offs[2 * b + 1], c);
    }
}

// Kernel 2: splat features at (g0-off0, g1-off1) directly into the sampled
// sub-grid of the output (residues always match pts_pick since g ≡ 0 mod 3).
__global__ __launch_bounds__(NTHREADS) void ksplat(const float* __restrict__ x,
                                                   const int* __restrict__ offs,
                                                   float* __restrict__ out)
{
    int b = blockIdx.x / CHUNKS, chunk = blockIdx.x % CHUNKS;
    const float* xb = x + (size_t)b * 6 * NPTS;
    float* ob = out + (size_t)b * 3 * GDIM * GDIM;
    int off0 = offs[2 * b], off1 = offs[2 * b + 1];
    int p0 = ((-off0) % 3 + 3) % 3;   // pts_pick, Python-style mod
    int p1 = ((-off1) % 3 + 3) % 3;
    int lane = threadIdx.x & 31, wave = threadIdx.x >> 5;
    for (int t = 0; t < GROUPS_PER_WAVE; ++t) {
        int n0 = chunk * PTS_PER_BLOCK + (wave * GROUPS_PER_WAVE + t) * 32;
        int n = n0 + lane;
        float e0, e1, e2;
        elevate_group(xb, n0, lane, e0, e1, e2);
        int gi0, gi1, r0, r1;
        lattice_point(e0, e1, e2, gi0, gi1, r0, r1);
        int c0 = gi0 - off0, c1 = gi1 - off1;   // coord >= 0 by construction
        if (c0 < S_DIM && c1 < S_DIM) {
            int i = (c0 - p0) / 3;   // exact: c0 ≡ p0 (mod 3)
            int j = (c1 - p1) / 3;
            float fe0 = xb[3 * NPTS + n];
            float fe1 = xb[4 * NPTS + n];
            float fe2 = xb[5 * NPTS + n];
            float* cell = ob + (size_t)i * GDIM + j;
            atomicAdd(cell, fe0);
            atomicAdd(cell + GDIM * GDIM, fe1);
            atomicAdd(cell + 2 * GDIM * GDIM, fe2);
        }
    }
}

extern "C" void kernel_launch(void* const* d_in, const int* in_sizes, int n_in,
                              void* d_out, int out_size, void* d_ws, size_t ws_size,
                              hipStream_t stream)
{
    (void)in_sizes; (void)n_in; (void)ws_size;
    const float* x = (const float*)d_in[0];     // (64, 6, 8192) f32
    float* out = (float*)d_out;                 // (64, 3, 128, 128) f32
    int* offs = (int*)d_ws;                     // 128 ints of scratch

    kinit<<<512, NTHREADS, 0, stream>>>((float4*)out, out_size / 4, offs);
    kmin<<<NBATCH * CHUNKS, NTHREADS, 0, stream>>>(x, offs);
    ksplat<<<NBATCH * CHUNKS, NTHREADS, 0, stream>>>(x, offs, out);
}